// WignerRotator_87436944212661
// MI455X (gfx1250) — compile-verified
//
#include <hip/hip_runtime.h>

typedef float v2f __attribute__((ext_vector_type(2)));
typedef float v8f __attribute__((ext_vector_type(8)));

#define NFW   2925
#define RROT  256
#define DFULL 169
#define TILES 28           // 8 (l=0..7: one 16x16 tile) + 5*4 (l=8..12: 2x2 tiles)
#define KSTEPS 7           // ceil(25/4): covers the largest block (d=25)

// flat offset of block l inside the 2925-wide flat coeff vector
__constant__ int c_off[13] = {0, 1, 10, 35, 84, 165, 286, 455, 680, 969, 1330, 1771, 2300};

__global__ __launch_bounds__(256)
void WignerRotator_wmma(const float* __restrict__ coeffs,   // (32, NFW)
                        const float* __restrict__ W,        // (256, 169, 169)
                        float* __restrict__ out)            // (32, 256, NFW)
{
    const int lane = threadIdx.x;                  // 0..31
    // Block-uniform work decode (pure scalar path):
    const int r    = blockIdx.x / TILES;
    const int tile = blockIdx.x - r * TILES;

    int l, ti, tj;
    if (tile < 8) { l = tile; ti = 0; tj = 0; }
    else { int t = tile - 8; l = 8 + (t >> 2); ti = (t >> 1) & 1; tj = t & 1; }

    const int d   = 2 * l + 1;
    const int s   = l * l;              // block start inside D=169
    const int off = c_off[l];           // block start inside flat nFW
    const int b0  = threadIdx.y * 4;    // 4 batches per wave

    // A fragment (W_l tile): lanes 0-15 -> K = {0,1}, lanes 16-31 -> K = {2,3}
    const int m        = lane & 15;
    const int kk       = (lane >> 4) * 2;
    const int arow     = ti * 16 + m;
    const int bcol     = tj * 16 + m;
    const bool arow_ok = (arow < d);
    const bool bcol_ok = (bcol < d);
    const int arow_c   = arow_ok ? arow : 0;   // clamped (always in-bounds)
    const int bcol_c   = bcol_ok ? bcol : 0;

    // 32-bit offsets from the scalar kernel-arg base pointers (saddr+voffset form)
    const int wbase = r * (DFULL * DFULL) + s * DFULL + s;   // W[r, s+i, s+k]
    const int cbase = b0 * NFW + off;                        // coeffs[b0, off + ...]

    v8f acc0 = {0.f,0.f,0.f,0.f,0.f,0.f,0.f,0.f};
    v8f acc1 = {0.f,0.f,0.f,0.f,0.f,0.f,0.f,0.f};
    v8f acc2 = {0.f,0.f,0.f,0.f,0.f,0.f,0.f,0.f};
    v8f acc3 = {0.f,0.f,0.f,0.f,0.f,0.f,0.f,0.f};

    #pragma unroll
    for (int kt = 0; kt < KSTEPS; ++kt) {
        const int  ka    = kt * 4 + kk;
        const int  kb    = ka + 1;
        const bool ka_ok = (ka < d);
        const bool kb_ok = (kb < d);
        const int  kac   = ka_ok ? ka : 0;     // clamped k indices
        const int  kbc   = kb_ok ? kb : 0;

        // A tile: unconditional in-bounds loads, zeroed by select afterwards
        const float a0 = W[wbase + arow_c * DFULL + kac];
        const float a1 = W[wbase + arow_c * DFULL + kbc];
        v2f a;
        a.x = (arow_ok && ka_ok) ? a0 : 0.f;
        a.y = (arow_ok && kb_ok) ? a1 : 0.f;

        // B tiles: unconditional loads into temps, then select (no branches)
        const int boffA = cbase + kac * d + bcol_c;
        const int boffB = cbase + kbc * d + bcol_c;
        const bool selA = (bcol_ok && ka_ok);
        const bool selB = (bcol_ok && kb_ok);

        const float t0A = coeffs[boffA + 0 * NFW];
        const float t0B = coeffs[boffB + 0 * NFW];
        const float t1A = coeffs[boffA + 1 * NFW];
        const float t1B = coeffs[boffB + 1 * NFW];
        const float t2A = coeffs[boffA + 2 * NFW];
        const float t2B = coeffs[boffB + 2 * NFW];
        const float t3A = coeffs[boffA + 3 * NFW];
        const float t3B = coeffs[boffB + 3 * NFW];

        v2f bf0, bf1, bf2, bf3;
        bf0.x = selA ? t0A : 0.f;  bf0.y = selB ? t0B : 0.f;
        bf1.x = selA ? t1A : 0.f;  bf1.y = selB ? t1B : 0.f;
        bf2.x = selA ? t2A : 0.f;  bf2.y = selB ? t2B : 0.f;
        bf3.x = selA ? t3A : 0.f;  bf3.y = selB ? t3B : 0.f;

        acc0 = __builtin_amdgcn_wmma_f32_16x16x4_f32(false, a, false, bf0, (short)0, acc0, false, false);
        acc1 = __builtin_amdgcn_wmma_f32_16x16x4_f32(false, a, false, bf1, (short)0, acc1, false, false);
        acc2 = __builtin_amdgcn_wmma_f32_16x16x4_f32(false, a, false, bf2, (short)0, acc2, false, false);
        acc3 = __builtin_amdgcn_wmma_f32_16x16x4_f32(false, a, false, bf3, (short)0, acc3, false, false);
    }

    // D layout (f32 16x16): VGPR v -> M = v + 8*(lane>=16), N = lane&15
    const int n     = lane & 15;
    const int col   = tj * 16 + n;
    const int mbase = (lane >> 4) * 8;

    // 32-bit output offsets off the scalar `out` base (max ~23.9M elements)
    const int obase = (b0 * RROT + r) * NFW + off;

    if (col < d) {
        #pragma unroll
        for (int v = 0; v < 8; ++v) {
            const int row = ti * 16 + mbase + v;
            if (row < d) {
                const int o = obase + row * d + col;
                out[o + 0 * (RROT * NFW)] = acc0[v];
                out[o + 1 * (RROT * NFW)] = acc1[v];
                out[o + 2 * (RROT * NFW)] = acc2[v];
                out[o + 3 * (RROT * NFW)] = acc3[v];
            }
        }
    }
}

extern "C" void kernel_launch(void* const* d_in, const int* in_sizes, int n_in,
                              void* d_out, int out_size, void* d_ws, size_t ws_size,
                              hipStream_t stream) {
    (void)in_sizes; (void)n_in; (void)d_ws; (void)ws_size; (void)out_size;
    const float* coeffs = (const float*)d_in[0];     // (32, 2925)
    const float* wig    = (const float*)d_in[1];     // (256, 169, 169)
    float* out          = (float*)d_out;             // (32, 256, 2925)

    dim3 block(32, 8);                               // 8 waves of 32
    dim3 grid(RROT * TILES);                         // 7168 blocks: (r, tile) per block
    hipLaunchKernelGGL(WignerRotator_wmma, grid, block, 0, stream, coeffs, wig, out);
}